// STPMLP_46488726012431
// MI455X (gfx1250) — compile-verified
//
#include <hip/hip_runtime.h>

// ---------------------------------------------------------------------------
// STPMLP on MI455X (gfx1250), compile-only target.
//
// Low-rank reformulation: stp_t = sum_{s<t} DECAY^(t-1-s) outer(left_s,right_s)
// so einsum(stp, x_t) = sum_s w_s * left_s * dot(right_s, x_t). The 67MB stp
// tensor is never materialized; the whole recurrence is 4 dependent skinny
// GEMMs/step ([32 x K] @ [K x N]) done with v_wmma_f32_16x16x4_f32 tiles.
//
// One persistent kernel, 16 workgroups x 256 threads (8 wave32 waves), one
// 16x16 output tile per wave (128 tiles per H=1024 GEMM), grid barrier via
// global atomics between stages. Activations staged row-major (stride K+4:
// 16B-aligned rows, conflict-free ds_load_b64 A-fragments) in LDS via
// GLOBAL_LOAD_ASYNC_TO_LDS_B128 (ASYNCcnt) when available. Weights stream
// from L2 with branchless global_prefetch_b8.
// ---------------------------------------------------------------------------

#define T_STEPS 16
#define B_N     32
#define I_N     512
#define H_N     1024
#define NBLK    16
#define TPB     256
#define NWAVE   (TPB / 32)
#define DECAY_F 0.999f

typedef __attribute__((ext_vector_type(2))) float v2f;
typedef __attribute__((ext_vector_type(8))) float v8f;
typedef __attribute__((ext_vector_type(4))) int   v4i;

#ifndef __has_builtin
#define __has_builtin(x) 0
#endif

#if __has_builtin(__builtin_amdgcn_global_load_async_to_lds_b128)
#define USE_ASYNC_LDS 1
#else
#define USE_ASYNC_LDS 0
#endif

// ---------------- grid-wide barrier (sense via monotonically growing gen) ---
__device__ inline void grid_barrier(unsigned* cnt, unsigned* gen) {
  __syncthreads();
  if (threadIdx.x == 0) {
    __threadfence();
    unsigned g = __hip_atomic_load(gen, __ATOMIC_RELAXED, __HIP_MEMORY_SCOPE_AGENT);
    unsigned arrived =
        __hip_atomic_fetch_add(cnt, 1u, __ATOMIC_ACQ_REL, __HIP_MEMORY_SCOPE_AGENT);
    if (arrived == NBLK - 1) {
      __hip_atomic_store(cnt, 0u, __ATOMIC_RELAXED, __HIP_MEMORY_SCOPE_AGENT);
      __hip_atomic_fetch_add(gen, 1u, __ATOMIC_RELEASE, __HIP_MEMORY_SCOPE_AGENT);
    } else {
      while (__hip_atomic_load(gen, __ATOMIC_ACQUIRE, __HIP_MEMORY_SCOPE_AGENT) == g) {
        __builtin_amdgcn_s_sleep(8);
      }
    }
  }
  __syncthreads();
}

// ---- stage Act[32][K] (row major, 16B aligned) into LDS rows of stride K+4 -
// Row stride K+4 floats: rows stay 16B aligned ((K+4)*4 % 16 == 0) and the
// 8-byte A-fragment reads (lanes m=0..15, address step (K+4) dwords -> bank
// step 4 on 64 banks) are conflict-free.
template <int K>
__device__ inline void load_act(float* __restrict__ ldsR, const float* __restrict__ act) {
  constexpr int SH = (K == 512) ? 9 : 10;
  constexpr int RS = K + 4;
  const int total = 32 * K;
  for (int i = threadIdx.x * 4; i < total; i += TPB * 4) {
    const int b = i >> SH;
    const int k = i & (K - 1);
#if USE_ASYNC_LDS
    // 16B contiguous in both global and LDS -> async DMA, no VGPR round-trip.
    // Param 1 is a flat v4i* (per clang's builtin signature); param 2 is the
    // LDS side: addrspace(3) offset = low 32 bits of the generic LDS address.
    __builtin_amdgcn_global_load_async_to_lds_b128(
        (v4i*)(uintptr_t)(act + i),
        (__attribute__((address_space(3))) v4i*)(unsigned)(uintptr_t)(ldsR + b * RS + k),
        0, 0);
#else
    const float4 v = *reinterpret_cast<const float4*>(act + i);
    float* r = ldsR + b * RS + k;
    r[0] = v.x; r[1] = v.y; r[2] = v.z; r[3] = v.w;
#endif
  }
#if USE_ASYNC_LDS
#if __has_builtin(__builtin_amdgcn_s_wait_asynccnt)
  __builtin_amdgcn_s_wait_asynccnt(0);
#else
  asm volatile("s_wait_asynccnt 0x0" ::: "memory");
#endif
#endif
}

// ---------------------------------------------------------------------------
// One 16x16 output tile of  Y[b0:b0+16, n0:n0+16] = Act @ Wop
//   TRB=true : Wop = W^T with W row-major [N][K]  (torch Linear layout)
//   TRB=false: Wop = W     with W row-major [K][N]
// A fragment (16x4 f32): lane L holds rows m=L&15, K-pair koff=2*(L>>4).
// B fragment (4x16 f32): lane L holds cols n=L&15, same K-pair (symmetric).
// Four interleaved accumulators break the WMMA dependence chain.
// ---------------------------------------------------------------------------
template <int K, bool TRB>
__device__ inline v8f gemm_tile(const float* __restrict__ ldsR,
                                const float* __restrict__ W, int ldw,
                                int n0, int b0, int lane) {
  constexpr int RS = K + 4;
  const int idx  = lane & 15;
  const int koff = (lane >> 4) << 1;
  const float* arow = ldsR + (size_t)(b0 + idx) * RS;
  const float* bptr = TRB ? (W + (size_t)(n0 + idx) * ldw) : (W + n0 + idx);

  v8f accs[4] = {v8f{}, v8f{}, v8f{}, v8f{}};
  for (int k = 0; k < K; k += 16) {
    if (TRB)  // branchless: wrap within the row (always a valid address)
      __builtin_prefetch(bptr + ((k + 128) & (K - 1)), 0, 1);
#pragma unroll
    for (int j = 0; j < 4; ++j) {
      const int ka = k + 4 * j + koff;
      const v2f a = *reinterpret_cast<const v2f*>(arow + ka);  // ds_load_b64
      v2f b;
      if (TRB) {
        b = *reinterpret_cast<const v2f*>(bptr + ka);          // global_load_b64
      } else {
        b.x = bptr[(size_t)(ka + 0) * ldw];
        b.y = bptr[(size_t)(ka + 1) * ldw];
      }
      accs[j] = __builtin_amdgcn_wmma_f32_16x16x4_f32(
          false, a, false, b, (short)0, accs[j], false, false);
    }
  }
  v8f acc;
#pragma unroll
  for (int i = 0; i < 8; ++i)
    acc[i] = (accs[0][i] + accs[1][i]) + (accs[2][i] + accs[3][i]);
  return acc;
}

// ---------------------------------------------------------------------------
__global__ __launch_bounds__(TPB, 1) void stpmlp_persistent(
    const float* __restrict__ x, const float* __restrict__ inpt_mat,
    const float* __restrict__ hid_mat, const float* __restrict__ main_mat,
    const float* __restrict__ main_bias, const float* __restrict__ W1,
    const float* __restrict__ b1, const float* __restrict__ W2,
    const float* __restrict__ b2, float* __restrict__ out,
    float* __restrict__ ws_left,   // [T][B][H]
    float* __restrict__ ws_right,  // [T][B][I]
    float* __restrict__ ws_y,      // [B][H]
    float* __restrict__ ws_h1,     // [B][H]
    float* __restrict__ ws_cmat,   // [T][T][B] : cmat[t][s][b] = dot(right_s[b], x_t[b])
    unsigned* cnt, unsigned* gen) {
  extern __shared__ float lds[];
  float* ldsA   = lds;                       // [32 * (H_N+4)] activation rows
  float* ldsX   = lds + 32 * (H_N + 4);      // [32 * (I_N+4)] x_t rows (stage D)
  float* c_last = ldsX + 32 * (I_N + 4);     // [B_N] freshest dot row

  const int tid  = threadIdx.x;
  const int lane = tid & 31;
  const int wave = tid >> 5;
  const int gw   = blockIdx.x * NWAVE + wave;  // 0..127 global wave id
  const int nidx = lane & 15;
  const int khi8 = (lane >> 4) << 3;           // C-layout: m = r + khi8
  const int n0   = (gw >> 1) << 4;             // N tile origin (H stages)
  const int b0   = (gw & 1) << 4;              // M tile origin

  for (int t = 0; t < T_STEPS; ++t) {
    const float* xt = x + (size_t)t * B_N * I_N;

    // ================= Stage A : y = x_t@main_mat^T + lowrank + bias =======
    load_act<I_N>(ldsA, xt);
    if (t >= 1 && tid < B_N) {  // redundant per-block: c_last[b] = dot(right_{t-1}[b], x_t[b])
      const float* rr = ws_right + ((size_t)(t - 1) * B_N + tid) * I_N;
      const float* xr = xt + (size_t)tid * I_N;
      float s = 0.f;
      for (int i = 0; i < I_N; i += 4)
        s += rr[i] * xr[i] + rr[i + 1] * xr[i + 1] + rr[i + 2] * xr[i + 2] +
             rr[i + 3] * xr[i + 3];
      c_last[tid] = s;
    }
    __syncthreads();
    {
      v8f acc = gemm_tile<I_N, true>(ldsA, main_mat, I_N, n0, b0, lane);
      const float bias = main_bias[n0 + nidx];
#pragma unroll
      for (int r = 0; r < 8; ++r) {
        const int m = r + khi8;
        float v = acc[r] + bias;
        float dp = 1.0f;  // DECAY^(t-1-s), s descending from t-1
        for (int s = t - 1; s >= 0; --s) {
          const float c = (s == t - 1)
                              ? c_last[b0 + m]
                              : ws_cmat[((size_t)t * T_STEPS + s) * B_N + (b0 + m)];
          v += dp * c * ws_left[((size_t)s * B_N + (b0 + m)) * H_N + n0 + nidx];
          dp *= DECAY_F;
        }
        ws_y[(size_t)(b0 + m) * H_N + n0 + nidx] = v;
      }
    }
    // cmat for strictly future steps: cmat[t'][t-1][b], t' >= t+1 (no race:
    // consumed only after >=4 grid barriers).
    if (t >= 1) {
      const int npairs = (T_STEPS - 1 - t) * B_N;
      const int g = blockIdx.x * TPB + tid;
      if (g < npairs) {
        const int tp = t + 1 + (g >> 5);
        const int b  = g & 31;
        const float* rr = ws_right + ((size_t)(t - 1) * B_N + b) * I_N;
        const float* xr = x + ((size_t)tp * B_N + b) * I_N;
        float s = 0.f;
        for (int i = 0; i < I_N; i += 4)
          s += rr[i] * xr[i] + rr[i + 1] * xr[i + 1] + rr[i + 2] * xr[i + 2] +
               rr[i + 3] * xr[i + 3];
        ws_cmat[((size_t)tp * T_STEPS + (t - 1)) * B_N + b] = s;
      }
    }
    grid_barrier(cnt, gen);

    // ================= Stage B : h1 = relu(y @ W1^T + b1) ===================
    load_act<H_N>(ldsA, ws_y);
    __syncthreads();
    {
      v8f acc = gemm_tile<H_N, true>(ldsA, W1, H_N, n0, b0, lane);
      const float bias = b1[n0 + nidx];
#pragma unroll
      for (int r = 0; r < 8; ++r) {
        const int m = r + khi8;
        ws_h1[(size_t)(b0 + m) * H_N + n0 + nidx] = fmaxf(acc[r] + bias, 0.f);
      }
    }
    grid_barrier(cnt, gen);

    // ================= Stage C : y_p = h1 @ W2^T + b2  (-> output) ==========
    load_act<H_N>(ldsA, ws_h1);
    __syncthreads();
    float* out_t = out + (size_t)t * B_N * H_N;
    {
      v8f acc = gemm_tile<H_N, true>(ldsA, W2, H_N, n0, b0, lane);
      const float bias = b2[n0 + nidx];
#pragma unroll
      for (int r = 0; r < 8; ++r) {
        const int m = r + khi8;
        out_t[(size_t)(b0 + m) * H_N + n0 + nidx] = acc[r] + bias;
      }
    }
    grid_barrier(cnt, gen);

    // ================= Stage D : left_t = y_p@hid^T ; right_t = x_t@inpt ====
    load_act<H_N>(ldsA, out_t);
    load_act<I_N>(ldsX, xt);
    __syncthreads();
    {
      v8f acc = gemm_tile<H_N, true>(ldsA, hid_mat, H_N, n0, b0, lane);
#pragma unroll
      for (int r = 0; r < 8; ++r) {
        const int m = r + khi8;
        ws_left[((size_t)t * B_N + b0 + m) * H_N + n0 + nidx] = acc[r];
      }
    }
    if (gw < 64) {  // 64 tiles for N = I = 512 (inpt_mat NOT transposed)
      const int jn0 = (gw >> 1) << 4;
      const int jb0 = (gw & 1) << 4;
      v8f acc = gemm_tile<I_N, false>(ldsX, inpt_mat, I_N, jn0, jb0, lane);
#pragma unroll
      for (int r = 0; r < 8; ++r) {
        const int m = r + khi8;
        ws_right[((size_t)t * B_N + jb0 + m) * I_N + jn0 + nidx] = acc[r];
      }
    }
    grid_barrier(cnt, gen);
  }
}

// ---------------------------------------------------------------------------
extern "C" void kernel_launch(void* const* d_in, const int* in_sizes, int n_in,
                              void* d_out, int out_size, void* d_ws, size_t ws_size,
                              hipStream_t stream) {
  (void)in_sizes; (void)n_in; (void)out_size; (void)ws_size;
  const float* x         = (const float*)d_in[0];
  const float* inpt_mat  = (const float*)d_in[1];
  const float* hid_mat   = (const float*)d_in[2];
  const float* main_mat  = (const float*)d_in[3];
  const float* main_bias = (const float*)d_in[4];
  const float* W1        = (const float*)d_in[5];
  const float* b1        = (const float*)d_in[6];
  const float* W2        = (const float*)d_in[7];
  const float* b2        = (const float*)d_in[8];
  float* out = (float*)d_out;

  // workspace carving (~3.3 MB): [sync 256B][left][right][y][h1][cmat]
  unsigned char* ws = (unsigned char*)d_ws;
  unsigned* cnt = (unsigned*)ws;
  unsigned* gen = (unsigned*)(ws + 64);
  float* f = (float*)(ws + 256);
  float* ws_left  = f; f += (size_t)T_STEPS * B_N * H_N;
  float* ws_right = f; f += (size_t)T_STEPS * B_N * I_N;
  float* ws_y     = f; f += (size_t)B_N * H_N;
  float* ws_h1    = f; f += (size_t)B_N * H_N;
  float* ws_cmat  = f;

  // zero barrier state every call (graph-capture safe, no cross-call state)
  (void)hipMemsetAsync(d_ws, 0, 256, stream);

  const size_t shmem =
      (size_t)(32 * (H_N + 4) + 32 * (I_N + 4) + 64) * sizeof(float);
  stpmlp_persistent<<<NBLK, TPB, shmem, stream>>>(
      x, inpt_mat, hid_mat, main_mat, main_bias, W1, b1, W2, b2, out,
      ws_left, ws_right, ws_y, ws_h1, ws_cmat, cnt, gen);
}